// AncProbsLayer_31877247271048
// MI455X (gfx1250) — compile-verified
//
#include <hip/hip_runtime.h>

typedef __attribute__((ext_vector_type(2))) float v2f;
typedef __attribute__((ext_vector_type(4))) float v4f;
typedef __attribute__((ext_vector_type(8))) float v8f;

#define MDIM 2
#define BDIM 32
#define LDIM 512
#define KDIM 8
#define SDIM 20
#define SPAD 32
#define NSWEEP 14

__device__ __forceinline__ float softplusf(float x) {
  // numerically stable log(1 + exp(x))
  return (x > 0.f) ? (x + log1pf(expf(-x))) : log1pf(expf(x));
}

// ---------------------------------------------------------------------------
// Phase 1: one block (1 wave) per (m,k). Build the normalized GTR generator,
// symmetrize with sqrt(p), run cyclic Jacobi to get lam/U, store padded to ws.
// ---------------------------------------------------------------------------
__global__ void __launch_bounds__(32)
anc_phase1(const float* __restrict__ exch,   // (m,k,S,S)
           const float* __restrict__ equil,  // (m,k,S)
           float* __restrict__ wsU,          // (m*k, 32, 32) padded eigenvectors
           float* __restrict__ wsLam,        // (m*k, 32)
           float* __restrict__ wsSq)         // (m*k, 32) sqrt(p), 1 in padding
{
  const int mk  = blockIdx.x;   // 0..15
  const int tid = threadIdx.x;

  __shared__ float Rm[SDIM][SDIM];
  __shared__ float Am[SDIM][SDIM];
  __shared__ float Um[SDIM][SDIM];
  __shared__ float pv[SDIM];
  __shared__ float sq[SDIM];
  __shared__ float rowsum[SDIM];
  __shared__ float muescale;

  const float* Kb = exch  + mk * SDIM * SDIM;
  const float* Eb = equil + mk * SDIM;

  // R = softplus((K + K^T)/2)
  for (int idx = tid; idx < SDIM * SDIM; idx += 32) {
    int i = idx / SDIM, j = idx - i * SDIM;
    Rm[i][j] = softplusf(0.5f * (Kb[i * SDIM + j] + Kb[j * SDIM + i]));
  }
  if (tid < SDIM) pv[tid] = Eb[tid];
  __syncthreads();

  // p = softmax(eq); sq = sqrt(p)   (tiny: serial on thread 0)
  if (tid == 0) {
    float mx = pv[0];
    for (int i = 1; i < SDIM; ++i) mx = fmaxf(mx, pv[i]);
    float sm = 0.f;
    for (int i = 0; i < SDIM; ++i) { float e = expf(pv[i] - mx); pv[i] = e; sm += e; }
    float inv = 1.f / sm;
    for (int i = 0; i < SDIM; ++i) { pv[i] *= inv; sq[i] = sqrtf(pv[i]); }
  }
  __syncthreads();

  // rowsum_i = sum_{j!=i} R_ij p_j   (= -Q_ii before normalization)
  if (tid < SDIM) {
    float s = 0.f;
    for (int j = 0; j < SDIM; ++j) if (j != tid) s += Rm[tid][j] * pv[j];
    rowsum[tid] = s;
  }
  __syncthreads();
  if (tid == 0) {
    float mue = 0.f;
    for (int i = 0; i < SDIM; ++i) mue += pv[i] * rowsum[i];
    muescale = 1.f / fmaxf(mue, 1e-16f);
  }
  __syncthreads();

  // Symmetrized generator: B_ij = R_ij sqrt(p_i p_j)/mue (i!=j), B_ii = -rowsum_i/mue
  for (int idx = tid; idx < SDIM * SDIM; idx += 32) {
    int i = idx / SDIM, j = idx - i * SDIM;
    Am[i][j] = (i == j) ? (-rowsum[i] * muescale)
                        : (Rm[i][j] * sq[i] * sq[j] * muescale);
    Um[i][j] = (i == j) ? 1.f : 0.f;
  }
  __syncthreads();

  // Cyclic Jacobi eigensolver: A <- J^T A J, U <- U J
  for (int sweep = 0; sweep < NSWEEP; ++sweep) {
    for (int p = 0; p < SDIM - 1; ++p) {
      for (int q = p + 1; q < SDIM; ++q) {
        float apq = Am[p][q];
        float c = 1.f, s = 0.f;
        if (fabsf(apq) > 1e-30f) {
          float theta = 0.5f * (Am[q][q] - Am[p][p]) / apq;
          float t = 1.f / (fabsf(theta) + sqrtf(theta * theta + 1.f));
          if (theta < 0.f) t = -t;
          c = 1.f / sqrtf(t * t + 1.f);
          s = t * c;
        }
        __syncthreads();
        if (tid < SDIM) {                 // rows p,q of J^T A
          float apt = Am[p][tid], aqt = Am[q][tid];
          Am[p][tid] = c * apt - s * aqt;
          Am[q][tid] = s * apt + c * aqt;
        }
        __syncthreads();
        if (tid < SDIM) {                 // cols p,q of (J^T A) J ; accumulate U J
          float atp = Am[tid][p], atq = Am[tid][q];
          Am[tid][p] = c * atp - s * atq;
          Am[tid][q] = s * atp + c * atq;
          float utp = Um[tid][p], utq = Um[tid][q];
          Um[tid][p] = c * utp - s * utq;
          Um[tid][q] = s * utp + c * utq;
        }
        __syncthreads();
      }
    }
  }

  float* Ub = wsU + mk * SPAD * SPAD;
  for (int idx = tid; idx < SPAD * SPAD; idx += 32) {
    int i = idx >> 5, j = idx & 31;
    Ub[idx] = (i < SDIM && j < SDIM) ? Um[i][j] : 0.f;
  }
  if (tid < SPAD) {
    wsLam[mk * SPAD + tid] = (tid < SDIM) ? Am[tid][tid] : 0.f;
    wsSq[mk * SPAD + tid]  = (tid < SDIM) ? sq[tid] : 1.f;
  }
}

// ---------------------------------------------------------------------------
// Phase 2: one block (4 waves) per (m,b,k).
//   P0 = (U * diag(exp(tau*lam))) x U^T  via f32 WMMA 16x16x4 (K = 20 = 5*4)
//   P  = P0 * sqrt(p_j)/sqrt(p_i), then gather rows by sequence letter using
//   16-byte non-temporal streaming stores (output is the 21 MB bottleneck).
// ---------------------------------------------------------------------------
__global__ void __launch_bounds__(128)
anc_phase2(const int*   __restrict__ seqs,   // (m,b,L)
           const int*   __restrict__ rates,  // (m,b)
           const float* __restrict__ tauk,   // (m,b)
           const float* __restrict__ wsU,
           const float* __restrict__ wsLam,
           const float* __restrict__ wsSq,
           float* __restrict__ out)          // (m,b,L,k,S)
{
  const int ki = blockIdx.x % KDIM;
  const int bi = (blockIdx.x / KDIM) % BDIM;
  const int mi = blockIdx.x / (KDIM * BDIM);
  const int mk = mi * KDIM + ki;
  const int tid  = threadIdx.x;
  const int lane = tid & 31;
  const int wave = tid >> 5;

  __shared__ __align__(16) float Up[SPAD][SPAD];
  __shared__ __align__(16) float Pp[SPAD][SPAD];
  __shared__ float ev[SPAD];
  __shared__ float sv[SPAD];
  __shared__ float rv[SPAD];
  __shared__ float taus;

  if (tid == 0) {
    int ri = rates[mi * BDIM + bi];
    taus = softplusf(tauk[mi * BDIM + ri]);
  }
  // padded U: 256 x 16B across 128 threads (global_load_b128 / ds_store_b128)
  const v4f* Ub4 = (const v4f*)(wsU + mk * SPAD * SPAD);
  v4f* Up4 = (v4f*)&Up[0][0];
  for (int idx = tid; idx < (SPAD * SPAD) / 4; idx += 128)
    Up4[idx] = Ub4[idx];
  __syncthreads();
  if (tid < SPAD) {
    float lam = wsLam[mk * SPAD + tid];
    ev[tid] = (tid < SDIM) ? expf(taus * lam) : 0.f;
    float s = wsSq[mk * SPAD + tid];
    sv[tid] = s;
    rv[tid] = 1.f / s;
  }
  __syncthreads();

  // Each wave owns one 16x16 tile of the padded 32x32 product.
  const int wi = wave >> 1, wj = wave & 1;
  const int Mb = wi * 16, Nb = wj * 16;
  const int lm = lane & 15;            // M (for A) / N (for B) within tile
  const int koff = (lane >> 4) * 2;    // upper half-wave covers K+2,K+3
  v8f acc = {0.f, 0.f, 0.f, 0.f, 0.f, 0.f, 0.f, 0.f};
  for (int kk = 0; kk < 5; ++kk) {     // K = 20 in 5 steps of 4
    int K0 = kk * 4 + koff;
    v2f a, bfr;
    a.x   = Up[Mb + lm][K0]     * ev[K0];      // A = U * diag(e), row-major frag
    a.y   = Up[Mb + lm][K0 + 1] * ev[K0 + 1];
    bfr.x = Up[Nb + lm][K0];                   // B[k][n] = U^T[k][n] = U[n][k]
    bfr.y = Up[Nb + lm][K0 + 1];
    acc = __builtin_amdgcn_wmma_f32_16x16x4_f32(
        /*neg_a=*/false, a, /*neg_b=*/false, bfr,
        /*c_mod=*/(short)0, acc, /*reuse_a=*/false, /*reuse_b=*/false);
  }

  // D layout: VGPR v -> M = v (+8 for lanes 16..31), N = lane%16.
  {
    const int hi8 = (lane >> 4) * 8;
    for (int v = 0; v < 8; ++v) {
      int M = Mb + hi8 + v;
      int N = Nb + lm;
      Pp[M][N] = acc[v] * sv[N] * rv[M];       // P_ij = M_ij * sqrt(p_j)/sqrt(p_i)
    }
  }
  __syncthreads();

  // Gather: out[m,b,l,ki,0:20] = P[seq[m,b,l]][0:20], as 5 16B chunks per l.
  // Row base offset l*K*S*4 = l*640 B and chunk offset c*16 B are 16B aligned.
  const int* seqb = seqs + (mi * BDIM + bi) * LDIM;
  float* outb = out + ((size_t)(mi * BDIM + bi) * LDIM * KDIM + ki) * SDIM;
  for (int idx = tid; idx < LDIM * (SDIM / 4); idx += 128) {
    int l = idx / (SDIM / 4), c = idx - l * (SDIM / 4);
    int aa = seqb[l];
    v4f val = *(const v4f*)&Pp[aa][c * 4];
    v4f* dst = (v4f*)(outb + (size_t)l * KDIM * SDIM + c * 4);
    __builtin_nontemporal_store(val, dst);     // streaming: don't pollute L2
  }
}

extern "C" void kernel_launch(void* const* d_in, const int* in_sizes, int n_in,
                              void* d_out, int out_size, void* d_ws, size_t ws_size,
                              hipStream_t stream) {
  (void)in_sizes; (void)n_in; (void)out_size; (void)ws_size;
  const int*   seqs  = (const int*)  d_in[0];  // sequences        (m,b,L) int32
  const int*   rates = (const int*)  d_in[1];  // rate_indices     (m,b)   int32
  const float* tauk  = (const float*)d_in[2];  // tau_kernel       (m,b)
  const float* exch  = (const float*)d_in[3];  // exchangeability  (m,k,S,S)
  const float* equil = (const float*)d_in[4];  // equilibrium      (m,k,S)
  float* out = (float*)d_out;

  float* ws    = (float*)d_ws;
  float* wsU   = ws;                                   // 16*32*32 floats
  float* wsLam = wsU + MDIM * KDIM * SPAD * SPAD;      // 16*32
  float* wsSq  = wsLam + MDIM * KDIM * SPAD;           // 16*32

  anc_phase1<<<MDIM * KDIM, 32, 0, stream>>>(exch, equil, wsU, wsLam, wsSq);
  anc_phase2<<<MDIM * BDIM * KDIM, 128, 0, stream>>>(seqs, rates, tauk,
                                                     wsU, wsLam, wsSq, out);
}